// GNN_53893249630545
// MI455X (gfx1250) — compile-verified
//
#include <hip/hip_runtime.h>

#define NN 50000
#define NE 800000
#define D 64

typedef float v2f __attribute__((ext_vector_type(2)));
typedef float v8f __attribute__((ext_vector_type(8)));

__device__ __forceinline__ void atomAddF(float* p, float v) {
    // native GLOBAL_ATOMIC_ADD_F32 on gfx1250
    unsafeAtomicAdd(p, v);
}

__global__ void s2v_zero(float* p, int n) {
    int i = blockIdx.x * blockDim.x + threadIdx.x;
    if (i < n) p[i] = 0.0f;
}

// u/v vectors: uv[0:64]=wfc1@relu(wvec1), uv[64:128]=wfc1@relu(-wvec1),
//              uv[128:192]=wfc2@relu(wvec2), uv[192:256]=wfc2@relu(-wvec2)
__global__ void s2v_prep_uv(const float* __restrict__ wfc1, const float* __restrict__ wvec1,
                            const float* __restrict__ wfc2, const float* __restrict__ wvec2,
                            float* __restrict__ uv) {
    int j = threadIdx.x;  // 64 threads
    float u1 = 0.f, v1 = 0.f, u2 = 0.f, v2 = 0.f;
    for (int k = 0; k < D; ++k) {
        float w1 = wvec1[k], w2 = wvec2[k];
        float a1 = wfc1[j * D + k], a2 = wfc2[j * D + k];
        u1 += a1 * fmaxf(w1, 0.f);
        v1 += a1 * fmaxf(-w1, 0.f);
        u2 += a2 * fmaxf(w2, 0.f);
        v2 += a2 * fmaxf(-w2, 0.f);
    }
    uv[j] = u1; uv[64 + j] = v1; uv[128 + j] = u2; uv[192 + j] = v2;
}

// P[d] += relu(w_e), Q[d] += relu(-w_e), hf1[d] += x[src]
__global__ void s2v_edge1(const float* __restrict__ w, const int* __restrict__ src,
                          const int* __restrict__ dst, const float* __restrict__ x,
                          float* __restrict__ P, float* __restrict__ Q,
                          float* __restrict__ hf1) {
    int e = blockIdx.x * blockDim.x + threadIdx.x;
    if (e >= NE) return;
    float wv = w[e];
    int d = dst[e], s = src[e];
    if (wv > 0.f)      atomAddF(&P[d], wv);
    else if (wv < 0.f) atomAddF(&Q[d], -wv);
    atomAddF(&hf1[2 * d + 0], x[2 * s + 0]);
    atomAddF(&hf1[2 * d + 1], x[2 * s + 1]);
}

// layer-1 node update: pure elementwise after the P/Q collapse
__global__ void s2v_node1(const float* __restrict__ x, const float* __restrict__ P,
                          const float* __restrict__ Q, const float* __restrict__ hf1,
                          const float* __restrict__ xfc1, const float* __restrict__ ffc1_w,
                          const float* __restrict__ ffc1_b, const float* __restrict__ uv,
                          float* __restrict__ h) {
    int idx = blockIdx.x * blockDim.x + threadIdx.x;
    if (idx >= NN * D) return;
    int n = idx >> 6, j = idx & 63;
    float acc = x[2 * n] * xfc1[2 * j] + x[2 * n + 1] * xfc1[2 * j + 1]
              + P[n] * uv[j] + Q[n] * uv[64 + j]
              + hf1[2 * n] * ffc1_w[2 * j] + hf1[2 * n + 1] * ffc1_w[2 * j + 1]
              + ffc1_b[j];
    h[idx] = fmaxf(acc, 0.f);
}

// one wave per edge, lane handles a contiguous float2 of the 64-wide row
__global__ void s2v_edge2(const int* __restrict__ src, const int* __restrict__ dst,
                          const float* __restrict__ h, float* __restrict__ hf2) {
    int t = blockIdx.x * blockDim.x + threadIdx.x;
    int e = t >> 5;
    int lane = t & 31;
    if (e >= NE) return;
    int s = src[e], d = dst[e];
    float2 hv = *(const float2*)&h[s * D + lane * 2];
    atomAddF(&hf2[d * D + lane * 2 + 0], hv.x);
    atomAddF(&hf2[d * D + lane * 2 + 1], hv.y);
}

// aggf2 = hf2 @ ffc2_w^T via V_WMMA_F32_16X16X4_F32, epilogue fused.
// wave -> one 16(M)x16(N) tile; 8 waves/block = 2 M-tiles x 4 N-tiles.
__global__ void __launch_bounds__(256) s2v_gemm_out(
        const float* __restrict__ hf2, const float* __restrict__ ffc2_w,
        const float* __restrict__ ffc2_b, const float* __restrict__ x,
        const float* __restrict__ P, const float* __restrict__ Q,
        const float* __restrict__ xfc2, const float* __restrict__ uv,
        float* __restrict__ out) {
    int wave = threadIdx.x >> 5;
    int lane = threadIdx.x & 31;
    int mtile = blockIdx.x * 2 + (wave >> 2);
    if (mtile >= NN / 16) return;   // wave-uniform exit; EXEC stays all-ones for WMMA
    int m0 = mtile * 16;
    int n0 = (wave & 3) * 16;

    int lhalf = lane >> 4;   // 0 | 1
    int lmod  = lane & 15;
    int n = n0 + lmod;

    v8f c = {};
    #pragma unroll
    for (int k0 = 0; k0 < D; k0 += 4) {
        int kb = k0 + (lhalf << 1);
        // A 16x4: lane m=lmod, holds K = kb, kb+1
        const float* ap = &hf2[(m0 + lmod) * D + kb];
        // B 4x16: B[k][n] = ffc2_w[n*64 + k] (ffc2_w is [out,in] row-major)
        const float* bp = &ffc2_w[n * D + kb];
        v2f a, b;
        a[0] = ap[0]; a[1] = ap[1];
        b[0] = bp[0]; b[1] = bp[1];
        c = __builtin_amdgcn_wmma_f32_16x16x4_f32(
                /*neg_a=*/false, a, /*neg_b=*/false, b,
                /*c_mod=*/(short)0, c, /*reuse_a=*/false, /*reuse_b=*/false);
    }

    float u2n = uv[128 + n], v2n = uv[192 + n], bn = ffc2_b[n];
    float xf0 = xfc2[2 * n], xf1 = xfc2[2 * n + 1];
    #pragma unroll
    for (int i = 0; i < 8; ++i) {
        int m = m0 + i + (lhalf << 3);   // C/D layout: VGPR i -> M=i (+8 for lanes 16-31)
        float acc = x[2 * m] * xf0 + x[2 * m + 1] * xf1
                  + P[m] * u2n + Q[m] * v2n + c[i] + bn;
        out[m * D + n] = fmaxf(acc, 0.f);
    }
}

extern "C" void kernel_launch(void* const* d_in, const int* in_sizes, int n_in,
                              void* d_out, int out_size, void* d_ws, size_t ws_size,
                              hipStream_t stream) {
    const float* x      = (const float*)d_in[0];
    const float* w      = (const float*)d_in[1];
    const int*   src    = (const int*)d_in[2];
    const int*   dst    = (const int*)d_in[3];
    const float* xfc1   = (const float*)d_in[4];
    const float* wfc1   = (const float*)d_in[5];
    const float* ffc1_w = (const float*)d_in[6];
    const float* ffc1_b = (const float*)d_in[7];
    const float* wvec1  = (const float*)d_in[8];
    const float* xfc2   = (const float*)d_in[9];
    const float* wfc2   = (const float*)d_in[10];
    const float* ffc2_w = (const float*)d_in[11];
    const float* ffc2_b = (const float*)d_in[12];
    const float* wvec2  = (const float*)d_in[13];
    float* out = (float*)d_out;

    float* ws  = (float*)d_ws;
    float* P   = ws;                         // [NN]
    float* Q   = ws + NN;                    // [NN]
    float* hf1 = ws + 2 * (size_t)NN;        // [NN*2]
    float* h   = ws + 4 * (size_t)NN;        // [NN*64]
    float* hf2 = ws + 68 * (size_t)NN;       // [NN*64]
    float* uv  = ws + 132 * (size_t)NN;      // [256]

    s2v_zero<<<(4 * NN + 255) / 256, 256, 0, stream>>>(P, 4 * NN);
    s2v_zero<<<(64 * NN + 255) / 256, 256, 0, stream>>>(hf2, 64 * NN);
    s2v_prep_uv<<<1, 64, 0, stream>>>(wfc1, wvec1, wfc2, wvec2, uv);
    s2v_edge1<<<(NE + 255) / 256, 256, 0, stream>>>(w, src, dst, x, P, Q, hf1);
    s2v_node1<<<(NN * D + 255) / 256, 256, 0, stream>>>(x, P, Q, hf1, xfc1, ffc1_w, ffc1_b, uv, h);
    s2v_edge2<<<(NE * 32 + 255) / 256, 256, 0, stream>>>(src, dst, h, hf2);
    s2v_gemm_out<<<(NN / 16 + 1) / 2, 256, 0, stream>>>(hf2, ffc2_w, ffc2_b, x, P, Q, xfc2, uv, out);
}